// OEQTensorProduct_70866960384262
// MI455X (gfx1250) — compile-verified
//
#include <hip/hip_runtime.h>

typedef __attribute__((ext_vector_type(2))) float v2f;
typedef __attribute__((ext_vector_type(8))) float v8f;

#define ROWS_PER_BLOCK 64
#define MID_STRIDE 68                          // padded: bank = (68*m + k)%64 = 4m+k -> conflict-free
#define MID_MAT (ROWS_PER_BLOCK * MID_STRIDE)  // 4352 floats per matrix
#define LDS_BYTES (4 * MID_MAT * 4)            // 69632 bytes

struct F3 { float a, b, c; };                  // 12-byte store => contiguous across lanes

__device__ __forceinline__ v8f wmma4(v2f a, v2f b, v8f c) {
  // D = A(16x4 f32) * B(4x16 f32) + C(16x16 f32)
  return __builtin_amdgcn_wmma_f32_16x16x4_f32(false, a, false, b, (short)0, c,
                                               false, false);
}

__global__ __launch_bounds__(256) void oeq_tp_kernel(
    const float* __restrict__ x,   // (Z,256)
    const float* __restrict__ y,   // (Z,4)
    const float* __restrict__ tp,  // (256,)
    const float* __restrict__ w0,  // (64,128)
    const float* __restrict__ w1,  // (64,128)
    float* __restrict__ out)       // (Z,512)
{
  extern __shared__ float smem[];
  float* midA = smem;                 // mid0 (for out0 GEMM)
  float* midK0 = smem + MID_MAT;      // mid1[:,:,0]
  float* midK1 = smem + 2 * MID_MAT;  // mid1[:,:,1]
  float* midK2 = smem + 3 * MID_MAT;  // mid1[:,:,2]

  const int t = threadIdx.x;
  const long rowBase = (long)blockIdx.x * ROWS_PER_BLOCK;

  // ---------------- Phase 1: elementwise mid -> LDS ----------------
  {
    const int m = t >> 2;   // local row 0..63
    const int q = t & 3;    // quarter of the u range
    const long row = rowBase + m;
    const float* xr = x + row * 256;
    const float* yr = y + row * 4;
    const float y0v = yr[0];
    const float ya = yr[1], yb = yr[2], yc = yr[3];
    const int u0 = q * 16;
#pragma unroll
    for (int j = 0; j < 16; ++j) {
      const int u = u0 + j;
      const float x0v = xr[u];
      const float xa = xr[64 + 3 * u + 0];
      const float xb = xr[64 + 3 * u + 1];
      const float xc = xr[64 + 3 * u + 2];
      const float wAv = tp[u];
      const float wBv = tp[64 + u];
      const float wCv = tp[128 + u];
      const float wDv = tp[192 + u];
      // scale 1/8 folded into mid (applies to both out0 and out1)
      const float m0 = (0.7071067811865476f * wAv) * x0v * y0v +
                       (0.4082482904638631f * wDv) * (xa * ya + xb * yb + xc * yc);
      const float bc = (0.7071067811865476f * wBv) * x0v;   // * y1[k]
      const float gc = (0.7071067811865476f * wCv) * y0v;   // * x1[u][k]
      midA[m * MID_STRIDE + u]  = m0 * 0.125f;
      midK0[m * MID_STRIDE + u] = (bc * ya + gc * xa) * 0.125f;
      midK1[m * MID_STRIDE + u] = (bc * yb + gc * xb) * 0.125f;
      midK2[m * MID_STRIDE + u] = (bc * yc + gc * xc) * 0.125f;
    }
  }
  __syncthreads();

  // ---------------- Phase 2: GEMMs via fp32 WMMA ----------------
  const int lane = t & 31;
  const int wv = t >> 5;       // wave 0..7
  const int l16 = lane & 15;   // M (A) / N (B,D) within tile
  const int lh = lane >> 4;    // 0 or 1 -> K offset +0 / +2

  // ---- out0: cols [16*wv, 16*wv+16), B from w0 ----
  {
    v2f B[16];
#pragma unroll
    for (int c = 0; c < 16; ++c) {
      const int k = 4 * c + 2 * lh;
      B[c].x = w0[k * 128 + wv * 16 + l16];
      B[c].y = w0[(k + 1) * 128 + wv * 16 + l16];
    }
#pragma unroll
    for (int s = 0; s < 4; ++s) {
      v8f acc = {};
      const float* Ap = midA + (s * 16 + l16) * MID_STRIDE + 2 * lh;
#pragma unroll
      for (int c = 0; c < 16; ++c) {
        const v2f a = *(const v2f*)(Ap + 4 * c);
        acc = wmma4(a, B[c], acc);
      }
      float* op = out + (rowBase + s * 16 + 8 * lh) * 512 + wv * 16 + l16;
#pragma unroll
      for (int v = 0; v < 8; ++v) op[(long)v * 512] = acc[v];
    }
  }

  // ---- out1: w-cols [16*wv, +16), 3 components share B from w1 ----
  {
    v2f B[16];
#pragma unroll
    for (int c = 0; c < 16; ++c) {
      const int k = 4 * c + 2 * lh;
      B[c].x = w1[k * 128 + wv * 16 + l16];
      B[c].y = w1[(k + 1) * 128 + wv * 16 + l16];
    }
#pragma unroll
    for (int s = 0; s < 4; ++s) {
      v8f acc0 = {}, acc1 = {}, acc2 = {};
      const int arow = (s * 16 + l16) * MID_STRIDE + 2 * lh;
      const float* A0 = midK0 + arow;
      const float* A1 = midK1 + arow;
      const float* A2 = midK2 + arow;
#pragma unroll
      for (int c = 0; c < 16; ++c) {
        const v2f a0 = *(const v2f*)(A0 + 4 * c);
        const v2f a1 = *(const v2f*)(A1 + 4 * c);
        const v2f a2 = *(const v2f*)(A2 + 4 * c);
        acc0 = wmma4(a0, B[c], acc0);
        acc1 = wmma4(a1, B[c], acc1);
        acc2 = wmma4(a2, B[c], acc2);
      }
      // output col = 128 + 3*wcol + k ; one lane owns all 3 k's of (row, wcol)
      float* op = out + (rowBase + s * 16 + 8 * lh) * 512 + 128 + 3 * (wv * 16 + l16);
#pragma unroll
      for (int v = 0; v < 8; ++v) {
        F3 f = {acc0[v], acc1[v], acc2[v]};
        *(F3*)(op + (long)v * 512) = f;
      }
    }
  }
}

extern "C" void kernel_launch(void* const* d_in, const int* in_sizes, int n_in,
                              void* d_out, int out_size, void* d_ws, size_t ws_size,
                              hipStream_t stream) {
  (void)in_sizes; (void)n_in; (void)d_ws; (void)ws_size; (void)out_size;
  const float* x = (const float*)d_in[0];
  const float* y = (const float*)d_in[1];
  const float* tp = (const float*)d_in[2];
  const float* w0 = (const float*)d_in[3];
  const float* w1 = (const float*)d_in[4];
  float* out = (float*)d_out;

  const int Z = 262144;
  dim3 grid(Z / ROWS_PER_BLOCK);  // 4096 blocks
  dim3 block(256);                // 8 wave32 waves
  oeq_tp_kernel<<<grid, block, LDS_BYTES, stream>>>(x, y, tp, w0, w1, out);
}